// VolumetricRenderer_52759378264704
// MI455X (gfx1250) — compile-verified
//
#include <hip/hip_runtime.h>
#include <math.h>

// NeRF volume rendering, MI455X (gfx1250, wave32).
// One wave per ray (S=192 = 32 lanes x 6 samples/lane).
//  - coalesced vector loads merged to global_load_b128/b64 th:TH_LOAD_NT
//    (all 354 MB is touched exactly once -> don't retain in cache)
//  - per-lane cumulative product + 5-step ds_bpermute multiplicative wave scan
//  - butterfly ds_bpermute reductions for rgb/depth/acc
//  - global_prefetch_b8 at WGP scope for the colors tile (consumed post-scan)
// Workload is HBM-bound (~354 MB @ 23.3 TB/s => ~15 us floor); it has no GEMM
// structure (cumprod is nonlinear, reductions are per-ray batched dots), so
// WMMA is intentionally not used -- it cannot move the bandwidth roofline.

typedef float f2 __attribute__((ext_vector_type(2)));

#define S_SAMPLES 192
#define CHUNK 6                 // samples per lane (32 * 6 = 192)
#define WPB 8                   // waves (rays) per block
#define BLOCK_T (WPB * 32)

__device__ __forceinline__ float lane_read_f32(float v, int src_lane) {
  // wave32: ds_bpermute uses addr bits [6:2] -> lanes 0..31
  return __int_as_float(
      __builtin_amdgcn_ds_bpermute(src_lane << 2, __float_as_int(v)));
}

__global__ __launch_bounds__(BLOCK_T)
void nerf_volrender_wave_scan(const float* __restrict__ dens,
                              const float* __restrict__ cols,
                              const float* __restrict__ tv,
                              const float* __restrict__ rdir,
                              float* __restrict__ rgb,
                              float* __restrict__ depth,
                              float* __restrict__ wout,
                              float* __restrict__ aout,
                              float* __restrict__ accw,
                              int R) {
  const int lane = (int)(threadIdx.x & 31u);
  const int ray  = (int)(blockIdx.x * WPB + (threadIdx.x >> 5));
  if (ray >= R) return;  // uniform per wave: EXEC stays all-ones below

  const size_t sbase = (size_t)ray * S_SAMPLES + (size_t)lane * CHUNK;
  const size_t cbase = (size_t)ray * (S_SAMPLES * 3) + (size_t)lane * (CHUNK * 3);

  // Kick colors toward the WGP early (global_prefetch_b8, WGP scope so it
  // fills all cache levels); the colors are consumed after the scan math.
  __builtin_prefetch(cols + cbase, 0, 3);

  // ray direction norm (12B broadcast, hot in L2)
  const float dx = rdir[3 * ray + 0];
  const float dy = rdir[3 * ray + 1];
  const float dz = rdir[3 * ray + 2];
  const float nrm = sqrtf(dx * dx + dy * dy + dz * dz);

  // t_vals and densities: 6 floats per lane, 8B-aligned, streamed (NT)
  const f2* tp = (const f2*)(tv + sbase);
  const f2* dp = (const f2*)(dens + sbase);
  const f2 t01 = __builtin_nontemporal_load(tp + 0);
  const f2 t23 = __builtin_nontemporal_load(tp + 1);
  const f2 t45 = __builtin_nontemporal_load(tp + 2);
  const f2 d01 = __builtin_nontemporal_load(dp + 0);
  const f2 d23 = __builtin_nontemporal_load(dp + 1);
  const f2 d45 = __builtin_nontemporal_load(dp + 2);

  float t[CHUNK] = { t01.x, t01.y, t23.x, t23.y, t45.x, t45.y };
  float d[CHUNK] = { d01.x, d01.y, d23.x, d23.y, d45.x, d45.y };

  // neighbor lane's first t for the chunk-boundary dist (lane 31 unused)
  const float tnext = lane_read_f32(t[0], lane + 1);

  float alpha[CHUNK], cum[CHUNK];
  float run = 1.0f;  // running product of p = (1 - alpha + 1e-10)
#pragma unroll
  for (int j = 0; j < CHUNK; ++j) {
    float dist;
    if (j < CHUNK - 1)      dist = t[j + 1] - t[j];
    else if (lane == 31)    dist = 1e10f;          // last sample -> infinity
    else                    dist = tnext - t[CHUNK - 1];
    const float x = fmaxf(d[j], 0.0f) * (dist * nrm);
    const float e = expf(-x);
    alpha[j] = 1.0f - e;
    run *= (e + 1e-10f);     // == 1 - alpha + 1e-10
    cum[j] = run;            // inclusive local cumprod
  }

  // Multiplicative exclusive scan of per-lane products across the wave.
  float scan = run;  // -> inclusive scan
#pragma unroll
  for (int off = 1; off < 32; off <<= 1) {
    const float v = lane_read_f32(scan, lane - off);
    if (lane >= off) scan *= v;
  }
  float texcl = lane_read_f32(scan, lane - 1);  // exclusive: prod over lanes < k
  if (lane == 0) texcl = 1.0f;

  // colors: 18 floats per lane (streamed once, NT; merges to b128s)
  const f2* cp = (const f2*)(cols + cbase);
  const f2 c0 = __builtin_nontemporal_load(cp + 0);
  const f2 c1 = __builtin_nontemporal_load(cp + 1);
  const f2 c2 = __builtin_nontemporal_load(cp + 2);
  const f2 c3 = __builtin_nontemporal_load(cp + 3);
  const f2 c4 = __builtin_nontemporal_load(cp + 4);
  const f2 c5 = __builtin_nontemporal_load(cp + 5);
  const f2 c6 = __builtin_nontemporal_load(cp + 6);
  const f2 c7 = __builtin_nontemporal_load(cp + 7);
  const f2 c8 = __builtin_nontemporal_load(cp + 8);
  const float col[CHUNK * 3] = { c0.x, c0.y, c1.x, c1.y, c2.x, c2.y,
                                 c3.x, c3.y, c4.x, c4.y, c5.x, c5.y,
                                 c6.x, c6.y, c7.x, c7.y, c8.x, c8.y };

  float w[CHUNK];
  float sr = 0.f, sg = 0.f, sb = 0.f, sd = 0.f, sw = 0.f;
#pragma unroll
  for (int j = 0; j < CHUNK; ++j) {
    const float trans = texcl * (j == 0 ? 1.0f : cum[j - 1]);
    const float wj = alpha[j] * trans;
    w[j] = wj;
    sr = fmaf(wj, col[3 * j + 0], sr);
    sg = fmaf(wj, col[3 * j + 1], sg);
    sb = fmaf(wj, col[3 * j + 2], sb);
    sd = fmaf(wj, t[j], sd);
    sw += wj;
  }

  // weights & alpha: 50 MB each, written once -> NT stores
  f2* wp = (f2*)(wout + sbase);
  f2* ap = (f2*)(aout + sbase);
  f2 w01; w01.x = w[0];     w01.y = w[1];
  f2 w23; w23.x = w[2];     w23.y = w[3];
  f2 w45; w45.x = w[4];     w45.y = w[5];
  f2 a01; a01.x = alpha[0]; a01.y = alpha[1];
  f2 a23; a23.x = alpha[2]; a23.y = alpha[3];
  f2 a45; a45.x = alpha[4]; a45.y = alpha[5];
  __builtin_nontemporal_store(w01, wp + 0);
  __builtin_nontemporal_store(w23, wp + 1);
  __builtin_nontemporal_store(w45, wp + 2);
  __builtin_nontemporal_store(a01, ap + 0);
  __builtin_nontemporal_store(a23, ap + 1);
  __builtin_nontemporal_store(a45, ap + 2);

  // Butterfly reductions across the wave (all lanes active).
#pragma unroll
  for (int off = 16; off > 0; off >>= 1) {
    sr += lane_read_f32(sr, lane ^ off);
    sg += lane_read_f32(sg, lane ^ off);
    sb += lane_read_f32(sb, lane ^ off);
    sd += lane_read_f32(sd, lane ^ off);
    sw += lane_read_f32(sw, lane ^ off);
  }
  if (lane == 0) {
    rgb[3 * (size_t)ray + 0] = sr;
    rgb[3 * (size_t)ray + 1] = sg;
    rgb[3 * (size_t)ray + 2] = sb;
    depth[ray] = sd;
    accw[ray]  = sw;
  }
}

extern "C" void kernel_launch(void* const* d_in, const int* in_sizes, int n_in,
                              void* d_out, int out_size, void* d_ws, size_t ws_size,
                              hipStream_t stream) {
  (void)n_in; (void)out_size; (void)d_ws; (void)ws_size;
  const float* dens = (const float*)d_in[0];  // [R, S]
  const float* cols = (const float*)d_in[1];  // [R, S, 3]
  const float* tv   = (const float*)d_in[2];  // [R, S]
  const float* rdir = (const float*)d_in[3];  // [R, 3]

  const int R = in_sizes[3] / 3;
  const size_t RS = (size_t)R * S_SAMPLES;

  float* out   = (float*)d_out;
  float* rgb   = out;                        // [R,3]
  float* depth = out + (size_t)3 * R;        // [R]
  float* wout  = out + (size_t)4 * R;        // [R,S]
  float* aout  = wout + RS;                  // [R,S]
  float* accw  = aout + RS;                  // [R]

  const int blocks = (R + WPB - 1) / WPB;
  hipLaunchKernelGGL(nerf_volrender_wave_scan, dim3(blocks), dim3(BLOCK_T), 0,
                     stream, dens, cols, tv, rdir, rgb, depth, wout, aout,
                     accw, R);
}